// UtilityLoss_73589969650309
// MI455X (gfx1250) — compile-verified
//
#include <hip/hip_runtime.h>
#include <stdint.h>

#define NDAYS 500
#define TILE  1024     // rows per block
#define TPB   256      // threads per block
#define RPT   4        // rows per thread (TILE = TPB * RPT)

__device__ __forceinline__ float fast_sigmoid12(float x) {
    // sigmoid(12x) = 1 / (1 + exp2(-12*log2(e)*x)); folds SCALING into one constant.
    const float k = -17.3123404906675611f; // -12 / ln(2)
#if __has_builtin(__builtin_amdgcn_exp2f)
    float e = __builtin_amdgcn_exp2f(k * x);
#else
    float e = exp2f(k * x);
#endif
    float denom = 1.0f + e;
#if __has_builtin(__builtin_amdgcn_rcpf)
    return __builtin_amdgcn_rcpf(denom);   // v_rcp_f32; overflow -> inf -> rcp = 0 (correct limit)
#else
    return 1.0f / denom;
#endif
}

// Streaming (single-use) global load with non-temporal cache hint (TH=NT).
__device__ __forceinline__ float nt_loadf(const float* p) {
    return __builtin_nontemporal_load(p);
}

__global__ void UtilityLoss_zero_kernel(float* __restrict__ Pi) {
    int i = threadIdx.x;
    if (i < NDAYS) Pi[i] = 0.0f;
}

__global__ void UtilityLoss_accum_kernel(const float* __restrict__ inputs,
                                         const float* __restrict__ targets,
                                         const float* __restrict__ weights,
                                         const int*   __restrict__ date,
                                         float* __restrict__ Pi,
                                         int N) {
    __shared__ float bins[NDAYS];
    __shared__ __align__(16) float wTile[TILE];
    __shared__ __align__(16) int   dTile[TILE];
    __shared__ int sRange[2];

    const int tid      = threadIdx.x;
    const int tileBase = (int)blockIdx.x * TILE;
    const int rowEnd   = (tileBase + TILE < N) ? (tileBase + TILE) : N;

    // --- Stage weights + date tiles into LDS via CDNA5 async global->LDS (b128 = 4 rows/unit)
    {
        const int r = tileBase + tid * 4;              // 16B-aligned global offset
        if (r + 4 <= N) {
            unsigned ldsW = (unsigned)(uintptr_t)&wTile[tid * 4];
            unsigned ldsD = (unsigned)(uintptr_t)&dTile[tid * 4];
            const float* gw = weights + r;
            const int*   gd = date    + r;
            asm volatile("global_load_async_to_lds_b128 %0, %1, off"
                         :: "v"(ldsW), "v"(gw) : "memory");
            asm volatile("global_load_async_to_lds_b128 %0, %1, off"
                         :: "v"(ldsD), "v"(gd) : "memory");
        }
    }

    // Zero segment bins while the async copies are in flight.
    for (int i = tid; i < NDAYS; i += TPB) bins[i] = 0.0f;

    // Dates are sorted: the block's day range is [date[first], date[last]].
    if (tid == 0) {
        sRange[0] = date[tileBase];
        sRange[1] = date[rowEnd - 1];
    }

#if __has_builtin(__builtin_amdgcn_s_wait_asynccnt)
    __builtin_amdgcn_s_wait_asynccnt(0);
#else
    asm volatile("s_wait_asynccnt 0" ::: "memory");
#endif
    __syncthreads();

    // --- Consume: 4 contiguous rows per thread, run-length accumulate by date.
    int   curD = -1;
    float acc  = 0.0f;
    const int r0 = tileBase + tid * RPT;
#pragma unroll
    for (int j = 0; j < RPT; ++j) {
        const int row = r0 + j;
        if (row < rowEnd) {
            const int li = row - tileBase;
            float w; int d;
            if ((row | 3) < N) {               // whole 4-row unit was staged
                w = wTile[li];
                d = dTile[li];
            } else {                           // unstaged tail (N % 4 rows)
                w = weights[row];
                d = date[row];
            }
            const float* ip = inputs  + row * 5;
            const float* tp = targets + row * 5;
            // Single-use stream: non-temporal loads keep WGP$/L2 clean.
            const float x0 = fast_sigmoid12(nt_loadf(ip + 0));
            const float x1 = fast_sigmoid12(nt_loadf(ip + 1));
            const float t0 = nt_loadf(tp + 0);
            const float t1 = nt_loadf(tp + 1);
            const float c  = w * (t0 * x0 + t1 * x1);
            if (d == curD) {
                acc += c;
            } else {
                if (curD >= 0) atomicAdd(&bins[curD], acc);   // ds_add_f32
                curD = d;
                acc  = c;
            }
        }
    }
    if (curD >= 0) atomicAdd(&bins[curD], acc);
    __syncthreads();

    // --- Flush only the block's (tiny, sorted) day range to global Pi.
    const int dmin = sRange[0], dmax = sRange[1];
    for (int d = dmin + tid; d <= dmax; d += TPB)
        atomicAdd(&Pi[d], bins[d]);
}

__global__ void UtilityLoss_finalize_kernel(const float* __restrict__ Pi,
                                            float* __restrict__ out) {
    __shared__ float s1[512];
    __shared__ float s2[512];
    const int t = threadIdx.x;
    const float p = (t < NDAYS) ? Pi[t] : 0.0f;
    s1[t] = p;
    s2[t] = p * p;
    __syncthreads();
#pragma unroll
    for (int off = 256; off > 0; off >>= 1) {
        if (t < off) { s1[t] += s1[t + off]; s2[t] += s2[t + off]; }
        __syncthreads();
    }
    if (t == 0) {
        const float s  = s1[0];
        const float q  = s2[0];
        const float sc = s > 0.0f ? s : 0.0f;          // clip(sumPi, 0, inf)
        out[0] = -(1.0f) * s * sc / q / (float)NDAYS;  // ALPHA = 1
    }
}

extern "C" void kernel_launch(void* const* d_in, const int* in_sizes, int n_in,
                              void* d_out, int out_size, void* d_ws, size_t ws_size,
                              hipStream_t stream) {
    const float* inputs  = (const float*)d_in[0];  // [N,5] f32
    const float* targets = (const float*)d_in[1];  // [N,5] f32
    const float* weights = (const float*)d_in[2];  // [N]   f32
    const int*   date    = (const int*)d_in[3];    // [N]   i32 (sorted)
    const int N = in_sizes[2];

    float* Pi = (float*)d_ws;                      // 500 floats of scratch

    UtilityLoss_zero_kernel<<<1, 512, 0, stream>>>(Pi);

    const int grid = (N + TILE - 1) / TILE;
    UtilityLoss_accum_kernel<<<grid, TPB, 0, stream>>>(inputs, targets, weights,
                                                       date, Pi, N);

    UtilityLoss_finalize_kernel<<<1, 512, 0, stream>>>(Pi, (float*)d_out);
}